// Attention_P_87960930222801
// MI455X (gfx1250) — compile-verified
//
#include <hip/hip_runtime.h>

typedef __attribute__((ext_vector_type(2))) float v2f;
typedef __attribute__((ext_vector_type(4))) float v4f;
typedef __attribute__((ext_vector_type(8))) float v8f;

#define BB  64
#define TK  2048
#define NN  512

// Hardware tanh (V_TANH_F32 is a gfx1250 TRANS op); fall back to libm.
__device__ __forceinline__ float fast_tanh(float x) {
#if __has_builtin(__builtin_amdgcn_tanhf)
  return __builtin_amdgcn_tanhf(x);
#else
  return tanhf(x);
#endif
}

// ---------------------------------------------------------------------------
// Kernel 1: dec_fea[64,512] = s_t_hat[64,512] @ W_dec^T[512,512] + b_dec
// One wave32 per 16x16 output tile, V_WMMA_F32_16X16X4_F32, K looped by 4.
// A layout (16x4 f32): lane holds M = lane&15, K = (lane>>4)*2 + {0,1}.
// B layout mirrors A with N in place of M:  B[k][n] = W_dec[n][k]  (W^T).
// C/D layout: VGPR r -> M = (lane>>4)*8 + r, N = lane&15.
// ---------------------------------------------------------------------------
__global__ __launch_bounds__(32)
void k_decproj_wmma(const float* __restrict__ s, const float* __restrict__ W,
                    const float* __restrict__ bias, float* __restrict__ dec) {
  const int lane  = threadIdx.x;
  const int tileM = blockIdx.x * 16;   // over b=64   -> gridDim.x = 4
  const int tileN = blockIdx.y * 16;   // over n=512  -> gridDim.y = 32
  const int mn = lane & 15;
  const int kq = (lane >> 4) * 2;      // 0 or 2

  const float* aRow = s + (tileM + mn) * NN + kq;  // A rows (s_t_hat)
  const float* bRow = W + (tileN + mn) * NN + kq;  // B cols = W rows (W^T)

  v8f acc = {};
  for (int kb = 0; kb < NN; kb += 4) {
    // 8-byte aligned contiguous K pair per lane -> global_load_b64
    v2f a = *(const v2f*)(aRow + kb);
    v2f b = *(const v2f*)(bRow + kb);
    acc = __builtin_amdgcn_wmma_f32_16x16x4_f32(
        /*neg_a=*/false, a, /*neg_b=*/false, b,
        /*c_mod=*/(short)0, acc, /*reuse_a=*/false, /*reuse_b=*/false);
  }

  const int row0 = tileM + (lane >> 4) * 8;
  const int col  = tileN + mn;
  const float bv = bias[col];
#pragma unroll
  for (int r = 0; r < 8; ++r)
    dec[(row0 + r) * NN + col] = acc[r] + bv;
}

// ---------------------------------------------------------------------------
// Kernel 2: scores[b,t] = sum_n tanh(enc_feat[b*TK+t, n] + dec_fea[b, n])*v[n]
// One wave32 per (b,t) row; b128 non-temporal loads on the 268 MB stream,
// hardware V_TANH_F32, wave32 butterfly reduction. HBM-bound.
// ---------------------------------------------------------------------------
__global__ __launch_bounds__(256)
void k_scores(const float* __restrict__ ef, const float* __restrict__ dec,
              const float* __restrict__ vp, float* __restrict__ scores) {
  const int wave = threadIdx.x >> 5;
  const int lane = threadIdx.x & 31;
  const long long row = (long long)blockIdx.x * 8 + wave;   // < BB*TK
  const int b = (int)(row >> 11);                           // row / TK

  const float* e = ef + row * NN;
  const float* d = dec + b * NN;

  float p = 0.f;
#pragma unroll
  for (int i0 = 0; i0 < NN; i0 += 128) {                    // 4 iterations
    const int idx = i0 + lane * 4;                          // 16B aligned
    v4f ev = __builtin_nontemporal_load((const v4f*)(e + idx));  // read-once
    v4f dv = *(const v4f*)(d + idx);                        // hot in cache
    v4f vv = *(const v4f*)(vp + idx);                       // hot in cache
    p += fast_tanh(ev.x + dv.x) * vv.x
       + fast_tanh(ev.y + dv.y) * vv.y
       + fast_tanh(ev.z + dv.z) * vv.z
       + fast_tanh(ev.w + dv.w) * vv.w;
  }

  // wave32 butterfly reduction
#pragma unroll
  for (int o = 16; o > 0; o >>= 1)
    p += __shfl_xor(p, o, 32);

  if (lane == 0) scores[row] = p;
}

// ---------------------------------------------------------------------------
// Kernel 3: masked renormalized softmax per batch row (t_k = 2048).
// softmax denom cancels: attn = exp(s-max)*mask / sum(exp(s-max)*mask)
// ---------------------------------------------------------------------------
__global__ __launch_bounds__(256)
void k_softmax(const float* __restrict__ scores, const float* __restrict__ mask,
               float* __restrict__ attn) {
  __shared__ float red[256];
  const int b = blockIdx.x, tid = threadIdx.x;
  const float* s = scores + b * TK;
  const float* mk = mask + b * TK;
  float* a = attn + b * TK;

  float m = -3.402823466e38f;
  for (int t = tid; t < TK; t += 256) m = fmaxf(m, s[t]);
  red[tid] = m; __syncthreads();
#pragma unroll
  for (int o = 128; o > 0; o >>= 1) {
    if (tid < o) red[tid] = fmaxf(red[tid], red[tid + o]);
    __syncthreads();
  }
  m = red[0]; __syncthreads();

  float sum = 0.f;
  for (int t = tid; t < TK; t += 256) {
    float w = __expf(s[t] - m) * mk[t];
    a[t] = w;
    sum += w;
  }
  red[tid] = sum; __syncthreads();
#pragma unroll
  for (int o = 128; o > 0; o >>= 1) {
    if (tid < o) red[tid] += red[tid + o];
    __syncthreads();
  }
  const float inv = 1.0f / red[0];

  for (int t = tid; t < TK; t += 256) a[t] *= inv;
}

// ---------------------------------------------------------------------------
// Kernel 4a: zero c_t (d_out is poisoned with 0xAA by the harness).
// ---------------------------------------------------------------------------
__global__ __launch_bounds__(256)
void k_zero(float* __restrict__ p, int nElem) {
  int i = blockIdx.x * 256 + threadIdx.x;
  if (i < nElem) p[i] = 0.f;
}

// ---------------------------------------------------------------------------
// Kernel 4b: c_t[b,n4] += sum_{t in seg} attn[b,t] * enc_out[b,t,n4]
// 128 threads x float4 cover all 512 columns; attn chunk staged in LDS;
// b128 non-temporal loads on the 268 MB stream; t split into 8 segments
// merged via global_atomic_add_f32.
// ---------------------------------------------------------------------------
__global__ __launch_bounds__(128)
void k_context(const float* __restrict__ attn, const float* __restrict__ enc,
               float* __restrict__ ct) {
  __shared__ float sa[256];
  const int b   = blockIdx.x;           // 64
  const int t0  = blockIdx.y * 256;     // 8 t-segments
  const int tid = threadIdx.x;          // 128 threads, 4 cols each

  sa[tid]       = attn[b * TK + t0 + tid];
  sa[tid + 128] = attn[b * TK + t0 + tid + 128];
  __syncthreads();

  const float* e = enc + ((long long)b * TK + t0) * NN + tid * 4;
  v4f acc = {0.f, 0.f, 0.f, 0.f};
#pragma unroll 4
  for (int t = 0; t < 256; ++t) {
    v4f ev = __builtin_nontemporal_load((const v4f*)(e + (long long)t * NN));
    acc += ev * sa[t];                  // v_fma on all 4 components
  }

  float* dst = ct + b * NN + tid * 4;
  atomicAdd(dst + 0, acc.x);
  atomicAdd(dst + 1, acc.y);
  atomicAdd(dst + 2, acc.z);
  atomicAdd(dst + 3, acc.w);
}

// ---------------------------------------------------------------------------
extern "C" void kernel_launch(void* const* d_in, const int* in_sizes, int n_in,
                              void* d_out, int out_size, void* d_ws, size_t ws_size,
                              hipStream_t stream) {
  const float* s_t_hat  = (const float*)d_in[0];   // [64, 512]
  const float* enc_out  = (const float*)d_in[1];   // [64, 2048, 512]
  const float* enc_feat = (const float*)d_in[2];   // [64*2048, 512]
  const float* mask     = (const float*)d_in[3];   // [64, 2048]
  const float* W_dec    = (const float*)d_in[4];   // [512, 512]
  const float* b_dec    = (const float*)d_in[5];   // [512]
  const float* v        = (const float*)d_in[6];   // [512]

  float* out  = (float*)d_out;
  float* ct   = out;               // c_t:   BB*NN  floats
  float* attn = out + BB * NN;     // attn:  BB*TK  floats

  float* dec    = (float*)d_ws;    // BB*NN floats   (128 KB)
  float* scores = dec + BB * NN;   // BB*TK floats   (512 KB)

  k_decproj_wmma<<<dim3(BB / 16, NN / 16), 32, 0, stream>>>(s_t_hat, W_dec, b_dec, dec);
  k_scores<<<(BB * TK) / 8, 256, 0, stream>>>(enc_feat, dec, v, scores);
  k_softmax<<<BB, 256, 0, stream>>>(scores, mask, attn);
  k_zero<<<(BB * NN) / 256, 256, 0, stream>>>(ct, BB * NN);
  k_context<<<dim3(BB, TK / 256), 128, 0, stream>>>(attn, enc_out, ct);
}